// Embedding_88227218195299
// MI455X (gfx1250) — compile-verified
//
#include <hip/hip_runtime.h>

// Embedding gather: out[b,s,:] = W[ids[b,s],:]
//   ids: [4,4096] int32, W: [50257,1024] f32, out: [16384,1024] f32
// Pure bandwidth problem: 64 MiB gathered read + 64 MiB write; no FLOPs, so no
// WMMA. Use the CDNA5 async LDS-DMA path (ASYNCcnt) so waves only issue
// descriptors; fall back to direct float4 copy if builtins are absent.
//
// Output stores are marked non-temporal (CPol TH=NT): the output is write-once
// and never re-read, while the 196 MB table nearly fits the 192 MB L2 — NT
// stores stop the output stream from evicting table rows.

#define THREADS      256   // 8 wave32 per block
#define ROW          1024  // floats per embedding row (4 KB)
#define TOK_PER_BLK  4     // rows staged per block (16 KB LDS, 4 async ops in flight/lane)

#define CPOL_RT      0     // default temporal
#define CPOL_NT      1     // TH[2:0]=NT, scope=default (gfx12-style cpol)

#if defined(__AMDGCN__) &&                                              \
    __has_builtin(__builtin_amdgcn_global_load_async_to_lds_b128) &&    \
    __has_builtin(__builtin_amdgcn_global_store_async_from_lds_b128)
#define USE_ASYNC_LDS 1
#else
#define USE_ASYNC_LDS 0
#endif

// b128 async builtins take pointers-to-int4 in AS1 (global) / AS3 (LDS):
//   void __builtin_amdgcn_global_load_async_to_lds_b128(v4i AS1*, v4i AS3*, imm offset, imm cpol)
typedef int v4i __attribute__((ext_vector_type(4)));
typedef __attribute__((address_space(1))) v4i g_v4i;  // global int4
typedef __attribute__((address_space(3))) v4i l_v4i;  // LDS int4

__device__ __forceinline__ void wait_async0() {
#if defined(__AMDGCN__)
#if __has_builtin(__builtin_amdgcn_s_wait_asynccnt)
  __builtin_amdgcn_s_wait_asynccnt(0);
#else
  asm volatile("s_wait_asynccnt 0" ::: "memory");
#endif
#endif
}

__global__ __launch_bounds__(THREADS) void embed_gather_kernel(
    const int* __restrict__ ids, const float* __restrict__ W,
    float* __restrict__ out, int n_tok) {
  const int tid  = threadIdx.x;
  const int tok0 = blockIdx.x * TOK_PER_BLK;

#if USE_ASYNC_LDS
  __shared__ float buf[TOK_PER_BLK * ROW];

  // Phase 1: issue async gathers, 4 B128 descriptors in flight per lane.
  // Each lane owns LDS bytes [16*tid, 16*tid+16) of its row slice -> no
  // cross-wave hazard, no barriers needed. Loads keep RT so W rows are
  // retained in L2.
#pragma unroll
  for (int t = 0; t < TOK_PER_BLK; ++t) {
    const int it = tok0 + t;
    if (it < n_tok) {
      const int tok = ids[it];  // block-uniform -> scalar load
      const float* src = W + (size_t)tok * ROW + (size_t)tid * 4;
      __builtin_amdgcn_global_load_async_to_lds_b128(
          (g_v4i*)src, (l_v4i*)(&buf[t * ROW + tid * 4]),
          /*offset=*/0, /*cpol=*/CPOL_RT);
    }
  }
  wait_async0();

  // Phase 2: stream rows out straight from LDS (no VGPR round trip),
  // non-temporal so the write stream doesn't evict table rows from L2.
#pragma unroll
  for (int t = 0; t < TOK_PER_BLK; ++t) {
    const int it = tok0 + t;
    if (it < n_tok) {
      float* dst = out + (size_t)it * ROW + (size_t)tid * 4;
      __builtin_amdgcn_global_store_async_from_lds_b128(
          (g_v4i*)dst, (l_v4i*)(&buf[t * ROW + tid * 4]),
          /*offset=*/0, /*cpol=*/CPOL_NT);
    }
  }
  wait_async0();
#else
  // Fallback: direct coalesced float4 copy (still bandwidth-optimal).
#pragma unroll
  for (int t = 0; t < TOK_PER_BLK; ++t) {
    const int it = tok0 + t;
    if (it < n_tok) {
      const int tok = ids[it];
      const float4* src = reinterpret_cast<const float4*>(W + (size_t)tok * ROW);
      float4* dst = reinterpret_cast<float4*>(out + (size_t)it * ROW);
      dst[tid] = src[tid];
    }
  }
#endif
}

extern "C" void kernel_launch(void* const* d_in, const int* in_sizes, int n_in,
                              void* d_out, int out_size, void* d_ws, size_t ws_size,
                              hipStream_t stream) {
  (void)n_in; (void)out_size; (void)d_ws; (void)ws_size;
  const int*   ids = (const int*)d_in[0];    // token_ids, [4*4096]
  const float* W   = (const float*)d_in[1];  // [50257*1024]
  float*       out = (float*)d_out;          // [16384*1024]

  const int n_tok  = in_sizes[0];            // 16384
  const int blocks = (n_tok + TOK_PER_BLK - 1) / TOK_PER_BLK;
  embed_gather_kernel<<<blocks, THREADS, 0, stream>>>(ids, W, out, n_tok);
}